// SAGENet_82016695484547
// MI455X (gfx1250) — compile-verified
//
#include <hip/hip_runtime.h>
#include <math.h>

#define N_NODES 50000
#define N_EDGES 800000
#define C_IN    100
#define HIDDEN  16
#define C_OUT   40

typedef __attribute__((ext_vector_type(2))) float v2f;
typedef __attribute__((ext_vector_type(8))) float v8f;

// ---------- zero fill ----------
__global__ __launch_bounds__(256) void k_zero(float* __restrict__ p, int n) {
    int t = blockIdx.x * 256 + threadIdx.x;
    if (t < n) p[t] = 0.0f;
}

// ---------- in-degree via fp32 atomics (resolved in L2; all buffers L2-resident) ----------
__global__ __launch_bounds__(256) void k_degree(const int* __restrict__ dst,
                                                float* __restrict__ deg) {
    int e = blockIdx.x * 256 + threadIdx.x;
    if (e < N_EDGES) atomicAdd(&deg[dst[e]], 1.0f);
}

// ---------- GEMM1: XW1 = X(50000x100) @ W1(100x16), fp32 WMMA 16x16x4 ----------
// One wave per 16-row tile. 3125 tiles total (50000/16 exact).
// A fragment (16x4 f32): lanes 0-15 -> M=lid, K = {k,k+1}; lanes 16-31 -> K = {k+2,k+3}.
// B fragment (4x16 f32): VGPR0 = row K (N across lanes), half-wave offset of 2 in K.
// C/D (16x16 f32): VGPR r -> M = r (+8 for upper half-wave), N = lid.
__global__ __launch_bounds__(256) void k_gemm1(const float* __restrict__ X,
                                               const float* __restrict__ W1,
                                               float* __restrict__ XW1) {
    const int lane = threadIdx.x & 31;
    const int tile = blockIdx.x * 8 + (threadIdx.x >> 5);
    if (tile * 16 >= N_NODES) return;          // wave-uniform: EXEC stays all-1s
    const int half = lane >> 4;                // 0 = lanes 0-15, 1 = lanes 16-31
    const int lid  = lane & 15;
    const float* xrow = X + (tile * 16 + lid) * C_IN;
    v8f c = {};
#pragma unroll
    for (int k = 0; k < C_IN; k += 4) {        // 25 WMMA ops, K=100 exact
        const int kk = k + half * 2;
        v2f a, b;
        a.x = xrow[kk];
        a.y = xrow[kk + 1];
        b.x = W1[(kk    ) * HIDDEN + lid];
        b.y = W1[(kk + 1) * HIDDEN + lid];
        c = __builtin_amdgcn_wmma_f32_16x16x4_f32(
                /*neg_a=*/false, a, /*neg_b=*/false, b,
                /*c_mod=*/(short)0, c, /*reuse_a=*/false, /*reuse_b=*/false);
    }
    float* orow = XW1 + tile * 16 * HIDDEN;
#pragma unroll
    for (int r = 0; r < 8; ++r)
        orow[(r + half * 8) * HIDDEN + lid] = c[r];
}

// ---------- edge aggregation: sum[dst] += msg[src], 16 channels as 4x float4 ----------
__global__ __launch_bounds__(256) void k_agg(const int* __restrict__ src,
                                             const int* __restrict__ dst,
                                             const float* __restrict__ msg,
                                             float* __restrict__ sum) {
    int t = blockIdx.x * 256 + threadIdx.x;
    if (t >= N_EDGES * 4) return;
    int e  = t >> 2;
    int c4 = (t & 3) * 4;
    int s = src[e], d = dst[e];
    const float4 v = *(const float4*)(msg + s * HIDDEN + c4);
    float* base = sum + d * HIDDEN + c4;
    atomicAdd(base + 0, v.x);
    atomicAdd(base + 1, v.y);
    atomicAdd(base + 2, v.z);
    atomicAdd(base + 3, v.w);
}

// ---------- h = relu(sum/deg + b1); also re-zeros sum for the layer-2 accumulation ----------
__global__ __launch_bounds__(256) void k_relu(float* __restrict__ sum,   // read then zeroed
                                              const float* __restrict__ deg,
                                              const float* __restrict__ b1,
                                              float* __restrict__ h) {
    int t = blockIdx.x * 256 + threadIdx.x;
    if (t >= N_NODES * HIDDEN) return;
    int node = t >> 4, c = t & 15;
    float sv = sum[t];
    float m  = sv / fmaxf(deg[node], 1.0f) + b1[c];
    h[t]   = fmaxf(m, 0.0f);
    sum[t] = 0.0f;                              // reuse buffer as sum2 accumulator
}

// ---------- out = log_softmax( (sum2/deg) @ W2 + b2 ), one node per thread ----------
__global__ __launch_bounds__(256) void k_out(const float* __restrict__ sum2,
                                             const float* __restrict__ deg,
                                             const float* __restrict__ W2,
                                             const float* __restrict__ b2,
                                             float* __restrict__ out) {
    int node = blockIdx.x * 256 + threadIdx.x;
    if (node >= N_NODES) return;
    float inv = 1.0f / fmaxf(deg[node], 1.0f);
    float m[HIDDEN];
#pragma unroll
    for (int k = 0; k < HIDDEN; ++k) m[k] = sum2[node * HIDDEN + k] * inv;
    float z[C_OUT];
    float zmax = -INFINITY;
#pragma unroll
    for (int j = 0; j < C_OUT; ++j) {
        float acc = b2[j];                      // W2/b2 indices lane-uniform -> SMEM loads
#pragma unroll
        for (int k = 0; k < HIDDEN; ++k) acc += m[k] * W2[k * C_OUT + j];
        z[j] = acc;
        zmax = fmaxf(zmax, acc);
    }
    float sexp = 0.0f;
#pragma unroll
    for (int j = 0; j < C_OUT; ++j) sexp += expf(z[j] - zmax);
    float lse = zmax + logf(sexp);
#pragma unroll
    for (int j = 0; j < C_OUT; ++j) out[node * C_OUT + j] = z[j] - lse;
}

extern "C" void kernel_launch(void* const* d_in, const int* in_sizes, int n_in,
                              void* d_out, int out_size, void* d_ws, size_t ws_size,
                              hipStream_t stream) {
    const float* x   = (const float*)d_in[0];            // [50000,100]
    const int*   ei  = (const int*)  d_in[1];            // [2,800000] row-major
    const float* W1  = (const float*)d_in[2];            // [100,16]
    const float* b1  = (const float*)d_in[3];            // [16]
    const float* W2  = (const float*)d_in[4];            // [16,40]
    const float* b2  = (const float*)d_in[5];            // [40]
    float* out = (float*)d_out;                          // [50000,40]

    const int* src = ei;
    const int* dst = ei + N_EDGES;

    // Workspace layout (floats): deg[50000] | bufA[800000] (XW1 then h) | bufB[800000] (sum1 then sum2)
    float* deg  = (float*)d_ws;
    float* bufA = deg  + N_NODES;
    float* bufB = bufA + N_NODES * HIDDEN;

    const int nz = N_NODES * HIDDEN;                     // 800000

    // zero accumulators (every call: graph replays don't re-poison)
    k_zero<<<(N_NODES + 255) / 256, 256, 0, stream>>>(deg, N_NODES);
    k_zero<<<(nz + 255) / 256, 256, 0, stream>>>(bufB, nz);

    // degrees
    k_degree<<<(N_EDGES + 255) / 256, 256, 0, stream>>>(dst, deg);

    // layer 1: GEMM first (linearity of mean), then 16-wide aggregation
    k_gemm1<<<(N_NODES / 16 + 7) / 8, 256, 0, stream>>>(x, W1, bufA);
    k_agg<<<(N_EDGES * 4 + 255) / 256, 256, 0, stream>>>(src, dst, bufA, bufB);
    k_relu<<<(nz + 255) / 256, 256, 0, stream>>>(bufB, deg, b1, bufA);

    // layer 2: aggregate 16-dim h, then 16->40 dense + log_softmax fused
    k_agg<<<(N_EDGES * 4 + 255) / 256, 256, 0, stream>>>(src, dst, bufA, bufB);
    k_out<<<(N_NODES + 255) / 256, 256, 0, stream>>>(bufB, deg, W2, b2, out);
}